// BicycleModel_48404281426003
// MI455X (gfx1250) — compile-verified
//
#include <hip/hip_runtime.h>
#include <hip/hip_bf16.h>
#include <stdint.h>

// Bicycle-model rollout, H=2048 steps x B=8192 rollouts.
// speed_t and yaw_t are closed-form in batch-independent prefix tables
// (S,K,C) computed once per block in LDS; only x,y need a per-rollout prefix,
// which is chunk-parallelized (8 chunks of 256 steps, two-pass).
#define H_STEPS 2048
#define BATCH   8192
#define NCH     8
#define CHL     (H_STEPS / NCH)   // 256
#define BLOCK   256
#define DT_F        0.05f
#define WHEELBASE_F 2.5f
#define MAX_STEER_F 0.5f
#define MAX_ACC_F   5.0f          // 5000/1000

typedef unsigned int v4u __attribute__((ext_vector_type(4)));
typedef int          v8i __attribute__((ext_vector_type(8)));
typedef int          v4i __attribute__((ext_vector_type(4)));

struct SharedMem {
  float a[H_STEPS];   // accel -> dv -> k*S temp
  float s[H_STEPS];   // steering -> k
  float S[H_STEPS];   // exclusive prefix of dv
  float K[H_STEPS];   // exclusive prefix of k
  float C[H_STEPS];   // exclusive prefix of k*S
  float tot[BLOCK];
  float partX[NCH][32];
  float partY[NCH][32];
};

#if defined(__gfx1250__) && __has_builtin(__builtin_amdgcn_tensor_load_to_lds) && \
    __has_builtin(__builtin_amdgcn_s_wait_tensorcnt)
#define USE_TDM 1
// Issue a TDM DMA of `nelem` fp32 from global to LDS (1-D tile). D# layout per
// CDNA5 ISA ch.8: group0 = {count=1 | lds_addr | global_addr | type=2},
// group1 = {data_size=4B, tensor_dim0=nelem, tensor_dim1=1, tile_dim0=nelem,
//           tile_dim1=1, tensor_dim0_stride=nelem}, groups 2/3 zero.
// This toolchain's builtin takes 6 operands (4 descriptor groups + an extra
// 8-dword group + cache policy); unused groups are zeroed.
__device__ __forceinline__ void tdm_load_f32(const void* gptr, void* ldsptr, int nelem) {
  unsigned long long ga = (unsigned long long)gptr;
  unsigned ldsoff = (unsigned)(unsigned long long)ldsptr;
  v4u g0 = { 1u,                                   // count=1, user descriptor
             ldsoff,                               // lds_addr
             (unsigned)(ga & 0xFFFFFFFFu),         // global_addr[31:0]
             (unsigned)((ga >> 32) & 0x1FFFFFFu) | (2u << 30) };  // addr[56:32] | type=2
  v8i g1;
  g1[0] = (int)(2u << 16);                                  // data_size = 4 bytes
  g1[1] = (int)(((unsigned)nelem & 0xFFFFu) << 16);         // tensor_dim0[15:0] @63:48
  g1[2] = (int)((((unsigned)nelem >> 16) & 0xFFFFu) | (1u << 16)); // dim0 hi | tensor_dim1=1
  g1[3] = (int)(((unsigned)nelem & 0xFFFFu) << 16);         // tile_dim0 @127:112
  g1[4] = 1;                                                // tile_dim1 = 1
  g1[5] = nelem;                                            // tensor_dim0_stride[31:0]
  g1[6] = 0;
  g1[7] = 0;
  v4i g2 = {0, 0, 0, 0};
  v4i g3 = {0, 0, 0, 0};
  v8i g4 = {0, 0, 0, 0, 0, 0, 0, 0};
  __builtin_amdgcn_tensor_load_to_lds(g0, g1, g2, g3, g4, 0);
}
#endif

// Block-level exclusive prefix scan of a 2048-element LDS array (256 threads:
// 8-wide serial segments + one wave32 shuffle scan over segment totals).
__device__ __forceinline__ void exscan2048(const float* src, float* dst, float* tot) {
  const int tid = threadIdx.x;
  float v[8];
  float run = 0.0f;
#pragma unroll
  for (int j = 0; j < 8; ++j) { v[j] = run; run += src[tid * 8 + j]; }
  tot[tid] = run;
  __syncthreads();
  if (tid < 32) {
    float lv[8];
    float lrun = 0.0f;
#pragma unroll
    for (int j = 0; j < 8; ++j) { lv[j] = lrun; lrun += tot[tid * 8 + j]; }
    float x = lrun;                       // wave32 inclusive scan via shfl_up
#pragma unroll
    for (int d = 1; d < 32; d <<= 1) {
      float n = __shfl_up(x, (unsigned)d, 32);
      if (tid >= d) x += n;
    }
    float laneExcl = x - lrun;            // exclusive across lane groups
#pragma unroll
    for (int j = 0; j < 8; ++j) tot[tid * 8 + j] = laneExcl + lv[j];
  }
  __syncthreads();
  float off = tot[tid];
#pragma unroll
  for (int j = 0; j < 8; ++j) dst[tid * 8 + j] = off + v[j];
  __syncthreads();
}

__global__ __launch_bounds__(BLOCK) void BicycleModel_rollout(
    const float* __restrict__ sx, const float* __restrict__ sy,
    const float* __restrict__ syaw, const float* __restrict__ ssp,
    const float* __restrict__ accel, const float* __restrict__ steer,
    float* __restrict__ out) {
  __shared__ SharedMem sm;
  const int tid = threadIdx.x;

  // ---- Stage the shared per-step control tables into LDS (TDM async DMA) ----
#ifdef USE_TDM
  if (tid < 32) {                         // one wave issues the DMA
    tdm_load_f32(accel, sm.a, H_STEPS);
    tdm_load_f32(steer, sm.s, H_STEPS);
    __builtin_amdgcn_s_wait_tensorcnt(0); // TENSORcnt == 0 -> data landed in LDS
  }
#else
  for (int j = tid; j < H_STEPS; j += BLOCK) { sm.a[j] = accel[j]; sm.s[j] = steer[j]; }
#endif
  __syncthreads();

  // ---- Per-step scalars: dv_t, k_t (batch-independent) ----
#pragma unroll
  for (int j = 0; j < 8; ++j) {
    const int t = tid * 8 + j;
    float a = sm.a[t];
    a = fminf(fmaxf(a, -1.0f), 1.0f);
    sm.a[t] = a * (DT_F * MAX_ACC_F);                 // dv_t
    float st = sm.s[t];
    st = fminf(fmaxf(st, -MAX_STEER_F), MAX_STEER_F);
    sm.s[t] = __tanf(st) * (DT_F / WHEELBASE_F);      // k_t
  }
  __syncthreads();

  // ---- Prefix tables: S = exscan(dv), K = exscan(k), C = exscan(k*S) ----
  exscan2048(sm.a, sm.S, sm.tot);
  exscan2048(sm.s, sm.K, sm.tot);
#pragma unroll
  for (int j = 0; j < 8; ++j) {
    const int t = tid * 8 + j;
    sm.a[t] = sm.s[t] * sm.S[t];
  }
  __syncthreads();
  exscan2048(sm.a, sm.C, sm.tot);

  // ---- Rollout: lanes = 32 consecutive batch elems, 8 time-chunks/rollout ----
  const int lane = tid & 31;
  const int c = tid >> 5;
  const int b = blockIdx.x * 32 + lane;
  const float x0 = sx[b], y0 = sy[b], yaw0 = syaw[b], sp0 = ssp[b];
  const int t0 = c * CHL;

  // Pass 1: chunk partial sums of speed_t*cos(yaw_t), speed_t*sin(yaw_t).
  float px = 0.0f, py = 0.0f;
#pragma unroll 4
  for (int t = t0; t < t0 + CHL; ++t) {
    const float sp = sp0 + sm.S[t];
    const float yaw = __fmaf_rn(sp0, sm.K[t], yaw0 + sm.C[t]);
    px = __fmaf_rn(sp, __cosf(yaw), px);
    py = __fmaf_rn(sp, __sinf(yaw), py);
  }
  sm.partX[c][lane] = px;
  sm.partY[c][lane] = py;
  __syncthreads();
  float ox = 0.0f, oy = 0.0f;
  for (int cc = 0; cc < c; ++cc) { ox += sm.partX[cc][lane]; oy += sm.partY[cc][lane]; }

  // Pass 2: emit all four state planes (coalesced 128B/wave/plane per step).
  float x = __fmaf_rn(DT_F, ox, x0);
  float y = __fmaf_rn(DT_F, oy, y0);
  float* __restrict__ outx   = out;
  float* __restrict__ outy   = out + (size_t)H_STEPS * BATCH;
  float* __restrict__ outyaw = out + 2 * (size_t)H_STEPS * BATCH;
  float* __restrict__ outsp  = out + 3 * (size_t)H_STEPS * BATCH;

#pragma unroll 4
  for (int t = t0; t < t0 + CHL; ++t) {
    const float sp = sp0 + sm.S[t];
    const float yaw = __fmaf_rn(sp0, sm.K[t], yaw0 + sm.C[t]);
    const size_t idx = (size_t)t * BATCH + b;
    outx[idx]   = x;
    outy[idx]   = y;
    outyaw[idx] = yaw;
    outsp[idx]  = sp;
    const float w = sp * DT_F;
    x = __fmaf_rn(w, __cosf(yaw), x);
    y = __fmaf_rn(w, __sinf(yaw), y);
  }
}

extern "C" void kernel_launch(void* const* d_in, const int* in_sizes, int n_in,
                              void* d_out, int out_size, void* d_ws, size_t ws_size,
                              hipStream_t stream) {
  (void)in_sizes; (void)n_in; (void)out_size; (void)d_ws; (void)ws_size;
  const float* sx    = (const float*)d_in[0];
  const float* sy    = (const float*)d_in[1];
  const float* syaw  = (const float*)d_in[2];
  const float* ssp   = (const float*)d_in[3];
  const float* accel = (const float*)d_in[4];
  const float* steer = (const float*)d_in[5];
  dim3 grid(BATCH / 32);   // 256 blocks x 256 threads = 2048 waves
  dim3 block(BLOCK);
  BicycleModel_rollout<<<grid, block, 0, stream>>>(sx, sy, syaw, ssp, accel, steer,
                                                   (float*)d_out);
}